// Nn0_86801289052729
// MI455X (gfx1250) — compile-verified
//
#include <hip/hip_runtime.h>
#include <math.h>

// ---- problem constants (from reference) ----
#define CTX   24     // C
#define DK    92
#define E     36     // EMBED_DIM
#define NTOK  500
#define HID   152
#define CCAT  936    // 2*E + E*CTX

// padded tile dims for 16x16x4 f32 WMMA
#define EP    48     // E padded to 3 M-tiles
#define DKP   96     // DK padded to 6 N-tiles

typedef __attribute__((ext_vector_type(2))) float v2f;
typedef __attribute__((ext_vector_type(8))) float v8f;

__device__ __forceinline__ v8f wmma_f32x4(v2f a, v2f b, v8f c) {
    // V_WMMA_F32_16X16X4_F32 : D = A(16x4,f32) * B(4x16,f32) + C(16x16,f32)
    return __builtin_amdgcn_wmma_f32_16x16x4_f32(
        /*neg_a=*/false, a, /*neg_b=*/false, b,
        /*c_mod=*/(short)0, c, /*reuse_a=*/false, /*reuse_b=*/false);
}

__global__ __launch_bounds__(512) void fused_attn_mlp(
    const float* __restrict__ x0,
    const float* __restrict__ wq1, const float* __restrict__ wk1, const float* __restrict__ wv1,
    const float* __restrict__ wq2, const float* __restrict__ wk2, const float* __restrict__ wv2,
    const float* __restrict__ nl0, const float* __restrict__ nl0_bias,
    const float* __restrict__ nl1,
    const float* __restrict__ ctp, const float* __restrict__ ctp_bias,
    float* __restrict__ out)
{
    __shared__ float x_pad[EP * CTX];      // x zero-padded to 48 rows
    __shared__ float qbuf[EP * DKP];       // q padded 48x96
    __shared__ float kbuf[EP * DKP];       // k padded 48x96
    __shared__ float sbuf[EP * EP];        // scores padded 48x48 (softmax in place)
    __shared__ float wvsum_s[CTX];
    __shared__ float vsum_s[E];
    __shared__ float t71_s[CCAT];
    __shared__ float t7_s[HID];
    __shared__ float t10_s[E];

    const int tid   = threadIdx.x;
    const int wave  = tid >> 5;
    const int lane  = tid & 31;
    const int lhalf = lane >> 4;   // 0: lanes 0-15, 1: lanes 16-31
    const int lmod  = lane & 15;

    // Warm the cache with nl0 (569 KB) while the WMMA attention phases run.
    // Lowers to global_prefetch_b8; no counter, no VGPR return.
    for (int off = tid * 64; off < CCAT * HID; off += 512 * 64)
        __builtin_prefetch(nl0 + off, 0, 1);

    // ---- init: stage x (zero-padded) + flattened x into t71 tail ----
    for (int i = tid; i < EP * CTX; i += 512) {
        int r = i / CTX, c = i % CTX;
        x_pad[i] = (r < E) ? x0[r * CTX + c] : 0.0f;
    }
    for (int i = tid; i < E * CTX; i += 512)
        t71_s[2 * E + i] = x0[i];
    __syncthreads();

    const float SCALE = 9.591663046625438f;  // sqrt(92)

    for (int head = 0; head < 2; ++head) {
        const float* wq = head ? wq2 : wq1;
        const float* wk = head ? wk2 : wk1;
        const float* wv = head ? wv2 : wv1;

        // wvsum[c] = sum_d wv[c,d]   (row sums of wv)
        if (tid < CTX) {
            float s = 0.0f;
            for (int d = 0; d < DK; ++d) s += wv[tid * DK + d];
            wvsum_s[tid] = s;
        }
        __syncthreads();
        // vsum[i] = (x @ wvsum)[i]  ==  rowsum(x @ wv)[i]
        if (tid < E) {
            float s = 0.0f;
            for (int c = 0; c < CTX; ++c) s += x_pad[tid * CTX + c] * wvsum_s[c];
            vsum_s[tid] = s;
        }

        // ---- phase 1: q = x@wq, k = x@wk via V_WMMA_F32_16X16X4_F32 ----
        // 2 matrices * 3 Mtiles * 6 Ntiles = 36 tile jobs, one wave each.
        for (int job = wave; job < 36; job += 16) {
            const float* w  = (job < 18) ? wq : wk;
            float*      dst = (job < 18) ? qbuf : kbuf;
            int rem = job % 18;
            int m0 = (rem / 6) * 16;
            int n0 = (rem % 6) * 16;
            v8f acc = {};
            #pragma unroll
            for (int kb = 0; kb < CTX; kb += 4) {
                // A fragment: 16x4 tile of x_pad
                int ak = kb + 2 * lhalf;
                int ar = m0 + lmod;
                v2f a; a[0] = x_pad[ar * CTX + ak];
                       a[1] = x_pad[ar * CTX + ak + 1];
                // B fragment: 4x16 tile of w (zero-pad cols >= DK)
                int bc = n0 + lmod;
                v2f b;
                b[0] = (bc < DK) ? w[(ak    ) * DK + bc] : 0.0f;
                b[1] = (bc < DK) ? w[(ak + 1) * DK + bc] : 0.0f;
                acc = wmma_f32x4(a, b, acc);
            }
            #pragma unroll
            for (int g = 0; g < 8; ++g)
                dst[(m0 + g + 8 * lhalf) * DKP + n0 + lmod] = acc[g];
        }
        __syncthreads();

        // ---- phase 2: s = q @ k^T via WMMA (K = 96 padded, zeros exact) ----
        for (int job = wave; job < 9; job += 16) {
            int m0 = (job / 3) * 16;
            int n0 = (job % 3) * 16;
            v8f acc = {};
            #pragma unroll
            for (int kb = 0; kb < DKP; kb += 4) {
                int ak = kb + 2 * lhalf;
                v2f a; a[0] = qbuf[(m0 + lmod) * DKP + ak];
                       a[1] = qbuf[(m0 + lmod) * DKP + ak + 1];
                // B = k^T  =>  B[kk][n] = kbuf[n][kk]
                int bn = n0 + lmod;
                v2f b; b[0] = kbuf[bn * DKP + ak];
                       b[1] = kbuf[bn * DKP + ak + 1];
                acc = wmma_f32x4(a, b, acc);
            }
            #pragma unroll
            for (int g = 0; g < 8; ++g)
                sbuf[(m0 + g + 8 * lhalf) * EP + n0 + lmod] = acc[g];
        }
        __syncthreads();

        // ---- phase 3: softmax over axis=0 (columns), scaled by sqrt(DK) ----
        if (tid < E) {
            int j = tid;
            float mx = -3.402823466e+38f;
            for (int i = 0; i < E; ++i)
                mx = fmaxf(mx, sbuf[i * EP + j] * SCALE);
            float sum = 0.0f;
            for (int i = 0; i < E; ++i) {
                float e = expf(sbuf[i * EP + j] * SCALE - mx);
                sbuf[i * EP + j] = e;
                sum += e;
            }
            float inv = 1.0f / sum;
            for (int i = 0; i < E; ++i)
                sbuf[i * EP + j] *= inv;
        }
        __syncthreads();

        // ---- phase 4: t[i] = sum_j scores[i,j] * vsum[j]  -> t71 slot ----
        if (tid < E) {
            float s = 0.0f;
            for (int j = 0; j < E; ++j) s += sbuf[tid * EP + j] * vsum_s[j];
            t71_s[head * E + tid] = s;
        }
        __syncthreads();
    }

    // ---- GEMV: t7 = relu(t71 @ nl0) + b0   (bandwidth-bound; coalesced) ----
    if (tid < HID) {
        float acc = 0.0f;
        for (int i = 0; i < CCAT; ++i)
            acc += t71_s[i] * nl0[i * HID + tid];    // lanes -> consecutive addrs
        t7_s[tid] = fmaxf(acc, 0.0f) + nl0_bias[tid];
    }
    __syncthreads();

    // ---- t10 = t7 @ nl1 ----
    if (tid < E) {
        float acc = 0.0f;
        for (int j = 0; j < HID; ++j) acc += t7_s[j] * nl1[j * E + tid];
        t10_s[tid] = acc;
    }
    __syncthreads();

    // ---- out = t10 @ ctp + ctp_bias ----
    if (tid < NTOK) {
        float acc = ctp_bias[tid];
        for (int e = 0; e < E; ++e) acc += t10_s[e] * ctp[e * NTOK + tid];
        out[tid] = acc;
    }
}

extern "C" void kernel_launch(void* const* d_in, const int* in_sizes, int n_in,
                              void* d_out, int out_size, void* d_ws, size_t ws_size,
                              hipStream_t stream) {
    (void)in_sizes; (void)n_in; (void)d_ws; (void)ws_size; (void)out_size;
    const float* x0       = (const float*)d_in[0];
    const float* wq1      = (const float*)d_in[1];
    const float* wk1      = (const float*)d_in[2];
    const float* wv1      = (const float*)d_in[3];
    const float* wq2      = (const float*)d_in[4];
    const float* wk2      = (const float*)d_in[5];
    const float* wv2      = (const float*)d_in[6];
    const float* nl0      = (const float*)d_in[7];
    const float* nl0_bias = (const float*)d_in[8];
    const float* nl1      = (const float*)d_in[9];
    const float* ctp      = (const float*)d_in[10];
    const float* ctp_bias = (const float*)d_in[11];
    float* out = (float*)d_out;

    fused_attn_mlp<<<1, 512, 0, stream>>>(x0, wq1, wk1, wv1, wq2, wk2, wv2,
                                          nl0, nl0_bias, nl1, ctp, ctp_bias, out);
}